// ReinforceLayer_39213051413051
// MI455X (gfx1250) — compile-verified
//
#include <hip/hip_runtime.h>
#include <math.h>

// Problem shape (matches reference setup_inputs).
constexpr int kN = 16384;   // N: points / row length
constexpr int kB = 512;     // batch
constexpr int TPB = 512;    // threads per block (16 waves, wave32)
constexpr float kSigmaBoost = 2.0f;
constexpr float kEpsilon = 1e-6f;

// ---------------- CDNA5 async global<->LDS helpers (probe-guarded) ----------------
#if defined(__gfx1250__) && __has_builtin(__builtin_amdgcn_global_load_async_to_lds_b128)
#define ASYNC_LOAD_LDS 1
#else
#define ASYNC_LOAD_LDS 0
#endif

#if defined(__gfx1250__) && __has_builtin(__builtin_amdgcn_global_store_async_from_lds_b128)
#define ASYNC_STORE_LDS 1
#else
#define ASYNC_STORE_LDS 0
#endif

typedef int v4i __attribute__((ext_vector_type(4)));
typedef __attribute__((address_space(1))) v4i* g_v4i_ptr;  // global
typedef __attribute__((address_space(3))) v4i* l_v4i_ptr;  // LDS

__device__ __forceinline__ void copy16_g2l(const float* __restrict__ g, float* l) {
#if ASYNC_LOAD_LDS
  // global_load_async_to_lds_b128: 16B/lane, tracked by ASYNCcnt
  __builtin_amdgcn_global_load_async_to_lds_b128(
      (g_v4i_ptr)(g), (l_v4i_ptr)(l), 0, 0);
#else
  *(float4*)l = *(const float4*)g;
#endif
}

__device__ __forceinline__ void copy16_l2g(float* __restrict__ g, const float* l) {
#if ASYNC_STORE_LDS
  __builtin_amdgcn_global_store_async_from_lds_b128(
      (g_v4i_ptr)(g), (l_v4i_ptr)(l), 0, 0);
#else
  *(float4*)g = *(const float4*)l;
#endif
}

__device__ __forceinline__ void wait_async0() {
#if ASYNC_LOAD_LDS || ASYNC_STORE_LDS
#if __has_builtin(__builtin_amdgcn_s_wait_asynccnt)
  __builtin_amdgcn_s_wait_asynccnt(0);
#else
  asm volatile("s_wait_asynccnt 0" ::: "memory");
#endif
#endif
}

// ---------------- Kernel 1: per-point hyper params (batch-independent) -------------
// params[t] = { sigmoid(pmeans[t,0])*(N-1),                 // .x = row mean
//               sigmoid(pmeans[t,1])*(N-1),                 // .y = col mean
//               (softplus(psigmas[t]+BOOST)+EPS)*N,         // .z = sigma
//               pvalues[t] }                                // .w = value
__global__ __launch_bounds__(256) void hyper_params_kernel(
    const float* __restrict__ pmeans, const float* __restrict__ psigmas,
    const float* __restrict__ pvalues, float4* __restrict__ params) {
  const int t = blockIdx.x * blockDim.x + threadIdx.x;
  if (t >= kN) return;
  const float scale = (float)(kN - 1);
  const float m0 = pmeans[2 * t + 0];
  const float m1 = pmeans[2 * t + 1];
  // numerically stable softplus: max(z,0) + log1p(exp(-|z|))
  const float z = psigmas[t] + kSigmaBoost;
  const float sp = fmaxf(z, 0.0f) + log1pf(expf(-fabsf(z)));
  float4 p;
  p.x = (1.0f / (1.0f + expf(-m0))) * scale;
  p.y = (1.0f / (1.0f + expf(-m1))) * scale;
  p.z = (sp + kEpsilon) * (float)kN;
  p.w = pvalues[t];
  params[t] = p;
}

// ---------------- Kernel 2: per-row gather + LDS scatter-add ----------------------
__global__ __launch_bounds__(TPB) void contract_kernel(
    const float* __restrict__ x, const float* __restrict__ noise,
    const float4* __restrict__ params, float* __restrict__ y) {
  __shared__ float xs[kN];  // 64 KB: staged x row (gather source)
  __shared__ float ys[kN];  // 64 KB: y row accumulator (scatter target)

  const int b = blockIdx.x;
  const int tid = threadIdx.x;
  const float* __restrict__ xrow = x + (size_t)b * kN;

  // Issue async copy of the x row into LDS (8 x b128 per thread, ASYNCcnt).
  for (int i = tid * 4; i < kN; i += TPB * 4) copy16_g2l(xrow + i, xs + i);

  // Overlap: zero the accumulator row with plain LDS stores.
  const float4 z4 = make_float4(0.f, 0.f, 0.f, 0.f);
  for (int i = tid * 4; i < kN; i += TPB * 4) *(float4*)(ys + i) = z4;

  wait_async0();
  __syncthreads();

  // noise row viewed as float4: one vector = noise pairs for points (2k, 2k+1).
  const float4* __restrict__ n4 = (const float4*)(noise + (size_t)b * kN * 2);
  const float hi = (float)(kN - 1);
#pragma unroll 4
  for (int k = tid; k < kN / 2; k += TPB) {  // (kN/2) % TPB == 0 -> no divergence
    const float4 nz = n4[k];            // b128: {n0a, n1a, n0b, n1b}
    const float4 pa = params[2 * k];    // b128: {mr, mc, sg, pv} point a
    const float4 pb = params[2 * k + 1];// b128: point b
    // round() under default RNE == jnp.round (half-to-even); clamp in float first.
    const float rfa = fminf(fmaxf(rintf(fmaf(pa.z, nz.x, pa.x)), 0.0f), hi);
    const float cfa = fminf(fmaxf(rintf(fmaf(pa.z, nz.y, pa.y)), 0.0f), hi);
    const float rfb = fminf(fmaxf(rintf(fmaf(pb.z, nz.z, pb.x)), 0.0f), hi);
    const float cfb = fminf(fmaxf(rintf(fmaf(pb.z, nz.w, pb.y)), 0.0f), hi);
    const float contribA = pa.w * xs[(int)cfa];  // LDS gather
    const float contribB = pb.w * xs[(int)cfb];
    atomicAdd(&ys[(int)rfa], contribA);          // ds_add_f32 (no return)
    atomicAdd(&ys[(int)rfb], contribB);
  }
  __syncthreads();

  // Stream the finished row back: async LDS->global stores.
  float* __restrict__ yrow = y + (size_t)b * kN;
  for (int i = tid * 4; i < kN; i += TPB * 4) copy16_l2g(yrow + i, ys + i);
  // ASYNCcnt drained by S_ENDPGM's implicit wait-idle.
}

// ---------------- Host-side launcher ----------------------------------------------
extern "C" void kernel_launch(void* const* d_in, const int* in_sizes, int n_in,
                              void* d_out, int out_size, void* d_ws, size_t ws_size,
                              hipStream_t stream) {
  const float* x       = (const float*)d_in[0];  // (B, N)
  const float* noise   = (const float*)d_in[1];  // (B, N, 2)
  const float* pmeans  = (const float*)d_in[2];  // (N, 2)
  const float* psigmas = (const float*)d_in[3];  // (N,)
  const float* pvalues = (const float*)d_in[4];  // (N,)
  float* y = (float*)d_out;                      // (B, N)

  float4* params = (float4*)d_ws;                // kN float4 = 256 KB scratch

  hyper_params_kernel<<<(kN + 255) / 256, 256, 0, stream>>>(pmeans, psigmas, pvalues, params);
  contract_kernel<<<kB, TPB, 0, stream>>>(x, noise, params, y);
}